// OutlierModel_25391846654132
// MI455X (gfx1250) — compile-verified
//
#include <hip/hip_runtime.h>
#include <hip/hip_bf16.h>

#define SIZE1 4096
#define SIZE2 4096
#define EMB   32
#define BQ    8192
#define KSEL  20
#define HID   64

typedef __attribute__((ext_vector_type(2))) float v2f;
typedef __attribute__((ext_vector_type(8))) float v8f;

// ---------------------------------------------------------------------------
// Kernel 1: per-row squared norms of emb1 and emb2 -> ws[0 .. 2*4096)
// ---------------------------------------------------------------------------
__global__ __launch_bounds__(256) void norms_kernel(const float* __restrict__ e1,
                                                    const float* __restrict__ e2,
                                                    float* __restrict__ ee) {
  int t = blockIdx.x * blockDim.x + threadIdx.x;
  if (t >= 2 * SIZE1) return;
  const float* e = (t < SIZE1) ? e1 : e2;
  int j = t & (SIZE1 - 1);
  const float* row = e + (size_t)j * EMB;
  float s = 0.f;
#pragma unroll
  for (int k = 0; k < EMB; ++k) s += row[k] * row[k];
  ee[t] = s;
}

// ---------------------------------------------------------------------------
// Kernel 2: KNN branch. One wave = 32 queries = two 16-row WMMA tiles.
// Distances D = Q @ E^T via V_WMMA_F32_16X16X4_F32 (K=32 in 8 steps).
// Rank by s[j] = ee[j] - 2*D (monotone-equivalent to w). Per-lane sorted
// top-20 in registers (self excluded). Then 20 exp/sqrt + 20 gathers.
// ---------------------------------------------------------------------------
__global__ __launch_bounds__(256) void knn_branch_kernel(
    const int* __restrict__ timep, const int* __restrict__ index1,
    const int* __restrict__ index2, const float* __restrict__ residuals,
    const float* __restrict__ emb1, const float* __restrict__ emb2,
    const float* __restrict__ ee_all, float* __restrict__ fout) {
  const int branch = blockIdx.y;                  // 0: emb1/index1, 1: emb2/index2
  const int lane = threadIdx.x & 31;
  const int wave = threadIdx.x >> 5;
  const int qbase = (blockIdx.x * 8 + wave) * 32;

  const float* emb  = branch ? emb2 : emb1;
  const int*  idxp  = branch ? index2 : index1;
  const int*  oidxp = branch ? index1 : index2;
  const float* ee   = ee_all + branch * SIZE1;

  __shared__ float lds[8][32 * 17];               // padded stride 17: conflict-free
  float* myLds = &lds[wave][0];

  const int h   = lane >> 4;                       // lane half (0/1)
  const int l15 = lane & 15;

  // this lane's own query
  const int b_own    = qbase + lane;
  const int idx_own  = idxp[b_own];
  const int oidx_own = oidxp[b_own];
  const int t_own    = timep[b_own];
  const float qq_own = ee[idx_own];                // ||q||^2 = ee[idx]

  // A fragments (16x4 fp32 tiles): lanes 0-15 rows M, halves hold K pairs {2h,2h+1}
  const int r0 = idxp[qbase + l15];
  const int r1 = idxp[qbase + 16 + l15];
  v2f a0[8], a1[8];
#pragma unroll
  for (int kk = 0; kk < 8; ++kk) {
    a0[kk] = *(const v2f*)(emb + (size_t)r0 * EMB + kk * 4 + h * 2);
    a1[kk] = *(const v2f*)(emb + (size_t)r1 * EMB + kk * 4 + h * 2);
  }

  // sorted (descending) top-20 smallest s values, register resident
  float ls[KSEL];
  int   lj[KSEL];
#pragma unroll
  for (int k = 0; k < KSEL; ++k) { ls[k] = 3.0e38f; lj[k] = 0; }

  for (int c = 0; c < SIZE1 / 16; ++c) {
    const int colbase = c * 16;
    v8f acc0 = {}, acc1 = {};
    const float* ecol = emb + (size_t)(colbase + l15) * EMB + h * 2;
#pragma unroll
    for (int kk = 0; kk < 8; ++kk) {
      v2f bb = *(const v2f*)(ecol + kk * 4);
      acc0 = __builtin_amdgcn_wmma_f32_16x16x4_f32(false, a0[kk], false, bb,
                                                   (short)0, acc0, false, false);
      acc1 = __builtin_amdgcn_wmma_f32_16x16x4_f32(false, a1[kk], false, bb,
                                                   (short)0, acc1, false, false);
    }
    const float eecol = ee[colbase + l15];
#pragma unroll
    for (int r = 0; r < 8; ++r) {
      myLds[(r + 8 * h) * 17 + l15]      = eecol - 2.0f * acc0[r];
      myLds[(16 + r + 8 * h) * 17 + l15] = eecol - 2.0f * acc1[r];
    }
    asm volatile("s_wait_dscnt 0" ::: "memory");

    // batch this lane's query row into registers (one wait, paired DS loads)
    const float* rowp = myLds + lane * 17;
    float sv[16];
#pragma unroll
    for (int i = 0; i < 16; ++i) sv[i] = rowp[i];

    // candidate loop runs purely from registers
#pragma unroll
    for (int i = 0; i < 16; ++i) {
      float s = sv[i];
      int j = colbase + i;
      if (j != idx_own && s < ls[0]) {
        ls[0] = s; lj[0] = j;                      // replace current max
#pragma unroll
        for (int k = 0; k < KSEL - 1; ++k) {       // one bubble pass restores order
          if (ls[k] < ls[k + 1]) {
            float ts = ls[k]; ls[k] = ls[k + 1]; ls[k + 1] = ts;
            int   tj = lj[k]; lj[k] = lj[k + 1]; lj[k + 1] = tj;
          }
        }
      }
    }
    asm volatile("" ::: "memory");                 // keep reads before next stores
  }

  // finalize: exp/sqrt only for the 20 winners; gather 20 residuals
  float wsum = 0.f, swsum = 0.f, ssum = 0.f, s2sum = 0.f;
  const float* resbase = residuals + (size_t)t_own * SIZE1 * SIZE2;
#pragma unroll
  for (int k = 0; k < KSEL; ++k) {
    float d2 = fmaxf(qq_own + ls[k], 0.0f);
    float w = expf(-(sqrtf(d2) + 0.001f));         // TAU = 1
    float sel = (branch == 0) ? resbase[(size_t)lj[k] * SIZE2 + oidx_own]
                              : resbase[(size_t)oidx_own * SIZE2 + lj[k]];
    wsum += w; swsum += w * sel; ssum += sel; s2sum += sel * sel;
  }
  float wmean = swsum / wsum;
  float msel = ssum * (1.0f / KSEL);
  float var = fmaxf((s2sum - (float)KSEL * msel * msel) * (1.0f / (KSEL - 1)), 0.0f);
  float* fo = fout + ((size_t)branch * BQ + b_own) * 3;
  fo[0] = wmean; fo[1] = wsum; fo[2] = sqrtf(var);
}

// ---------------------------------------------------------------------------
// Kernel 3: 8 -> 64 (relu) -> {mean_outlier, std_outlier}
// ---------------------------------------------------------------------------
__global__ __launch_bounds__(256) void mlp_kernel(
    const int* __restrict__ timep, const int* __restrict__ index1,
    const int* __restrict__ index2, const float* __restrict__ means,
    const float* __restrict__ stds, const float* __restrict__ W1,
    const float* __restrict__ b1, const float* __restrict__ Wm,
    const float* __restrict__ bm, const float* __restrict__ Ws,
    const float* __restrict__ bs, const float* __restrict__ fbuf,
    float* __restrict__ out) {
  int b = blockIdx.x * blockDim.x + threadIdx.x;
  if (b >= BQ) return;
  int t = timep[b], i1 = index1[b], i2 = index2[b];
  size_t off = ((size_t)t * SIZE1 + i1) * SIZE2 + i2;
  const float* f1 = fbuf + (size_t)b * 3;
  const float* f2 = fbuf + (size_t)(BQ + b) * 3;
  float feats[8] = {f1[0], f1[1], f1[2], f2[0], f2[1], f2[2], means[off], stds[off]};
  float mo = bm[0], so = bs[0];
  for (int j = 0; j < HID; ++j) {
    float hv = b1[j];
#pragma unroll
    for (int k = 0; k < 8; ++k) hv += W1[j * 8 + k] * feats[k];
    hv = fmaxf(hv, 0.0f);
    mo += Wm[j] * hv;
    so += Ws[j] * hv;
  }
  out[b] = mo;
  out[BQ + b] = so;
}

// ---------------------------------------------------------------------------
extern "C" void kernel_launch(void* const* d_in, const int* in_sizes, int n_in,
                              void* d_out, int out_size, void* d_ws, size_t ws_size,
                              hipStream_t stream) {
  const int*   timep     = (const int*)d_in[0];
  const int*   index1    = (const int*)d_in[1];
  const int*   index2    = (const int*)d_in[2];
  const float* residuals = (const float*)d_in[3];
  const float* means     = (const float*)d_in[4];
  const float* stds      = (const float*)d_in[5];
  const float* emb1      = (const float*)d_in[6];
  const float* emb2      = (const float*)d_in[7];
  const float* W1        = (const float*)d_in[8];
  const float* b1        = (const float*)d_in[9];
  const float* Wm        = (const float*)d_in[10];
  const float* bm        = (const float*)d_in[11];
  const float* Ws        = (const float*)d_in[12];
  const float* bs        = (const float*)d_in[13];

  float* ee   = (float*)d_ws;              // [2][4096]
  float* fbuf = ee + 2 * SIZE1;            // [2][BQ][3]
  float* out  = (float*)d_out;             // [2][BQ]

  norms_kernel<<<(2 * SIZE1 + 255) / 256, 256, 0, stream>>>(emb1, emb2, ee);

  dim3 grid(BQ / 256, 2);                  // 32 query-blocks x 2 branches
  knn_branch_kernel<<<grid, 256, 0, stream>>>(timep, index1, index2, residuals,
                                              emb1, emb2, ee, fbuf);

  mlp_kernel<<<(BQ + 255) / 256, 256, 0, stream>>>(timep, index1, index2, means,
                                                   stds, W1, b1, Wm, bm, Ws, bs,
                                                   fbuf, out);
}